// QKVParallelConvolution_4612794876168
// MI455X (gfx1250) — compile-verified
//
#include <hip/hip_runtime.h>
#include <cstdint>

#ifndef __has_builtin
#define __has_builtin(x) 0
#endif

// Problem sizes (fixed by the reference)
#define BB 4
#define LL 2048
#define DD 8192
#define KK 4

// Tiling
#define DBLK  1024            // channels per block (256 threads x float4)
#define TLL   8               // timesteps per LDS tile
#define NTILE 8               // tiles per block
#define LPB   (TLL * NTILE)   // 64 rows per block
#define LSPLIT (LL / LPB)     // 32 row-spans

typedef float v4f __attribute__((ext_vector_type(4)));
typedef int   v4i __attribute__((ext_vector_type(4)));

typedef __attribute__((address_space(1))) v4i* gbl_v4i_ptr;  // printed as "int4 __device__ *"
typedef __attribute__((address_space(3))) v4i* lds_v4i_ptr;  // printed as "int4 __shared__ *"

// ---- CDNA5 async global->LDS copy (16B per lane), ASYNCcnt-tracked ----
__device__ __forceinline__ void async_copy_b128(const float* g, float* l) {
#if __has_builtin(__builtin_amdgcn_global_load_async_to_lds_b128)
  __builtin_amdgcn_global_load_async_to_lds_b128(
      (gbl_v4i_ptr)g, (lds_v4i_ptr)l, /*imm offset=*/0, /*cpol=*/0);
#else
  unsigned lds32 = (unsigned)(uintptr_t)(lds_v4i_ptr)l;
  unsigned long long ga = (unsigned long long)(uintptr_t)g;
  asm volatile("global_load_async_to_lds_b128 %0, %1, off"
               :: "v"(lds32), "v"(ga) : "memory");
#endif
}

__device__ __forceinline__ void wait_async_all() {
#if __has_builtin(__builtin_amdgcn_s_wait_asynccnt)
  __builtin_amdgcn_s_wait_asynccnt(0);
#else
  asm volatile("s_wait_asynccnt 0" ::: "memory");
#endif
}

__device__ __forceinline__ void wait_async_prev() {   // leave newest TLL=8 in flight
#if __has_builtin(__builtin_amdgcn_s_wait_asynccnt)
  __builtin_amdgcn_s_wait_asynccnt(8);
#else
  asm volatile("s_wait_asynccnt 8" ::: "memory");
#endif
}

__device__ __forceinline__ float silu1(float v) {
  return v / (1.0f + __expf(-v));
}

// y[t] = w0*x[t-3] + w1*x[t-2] + w2*x[t-1] + w3*x[t]; silu; split q/k/v.
__global__ __launch_bounds__(256) void conv_silu_kernel(
    const float* __restrict__ x, const float* __restrict__ cache,
    const float* __restrict__ weight, float* __restrict__ out) {
  __shared__ float smem[2 * TLL * DBLK];   // 64 KB, double buffered

  const int tid = threadIdx.x;
  const int nd = DD / DBLK;                // 8
  int blk = blockIdx.x;
  const int b  = blk / (nd * LSPLIT);
  int r = blk % (nd * LSPLIT);
  const int dt = r / LSPLIT;
  const int lt = r % LSPLIT;
  const int d0 = dt * DBLK;
  const int l0 = lt * LPB;
  const int c  = d0 + tid * 4;             // first of this thread's 4 channels

  // Per-channel taps: weight[(c+i)*4 + k], k ascending = oldest..newest
  const v4f w0 = *(const v4f*)(weight + (size_t)(c + 0) * KK);
  const v4f w1 = *(const v4f*)(weight + (size_t)(c + 1) * KK);
  const v4f w2 = *(const v4f*)(weight + (size_t)(c + 2) * KK);
  const v4f w3 = *(const v4f*)(weight + (size_t)(c + 3) * KK);

  // Sliding causal window (registers): values at t-3, t-2, t-1
  const float* xb = x + (size_t)b * LL * DD + c;   // row base for this thread
  v4f xm3, xm2, xm1;
  if (l0 == 0) {
    // cache[b, ch, k]: slot 1 -> t-3, slot 2 -> t-2, slot 3 -> t-1
    const float* cb = cache + ((size_t)b * DD + c) * KK;
    v4f c0 = *(const v4f*)(cb + 0 * KK);
    v4f c1 = *(const v4f*)(cb + 1 * KK);
    v4f c2 = *(const v4f*)(cb + 2 * KK);
    v4f c3 = *(const v4f*)(cb + 3 * KK);
    xm3 = (v4f){c0.y, c1.y, c2.y, c3.y};
    xm2 = (v4f){c0.z, c1.z, c2.z, c3.z};
    xm1 = (v4f){c0.w, c1.w, c2.w, c3.w};
  } else {
    xm3 = *(const v4f*)(xb + (size_t)(l0 - 3) * DD);
    xm2 = *(const v4f*)(xb + (size_t)(l0 - 2) * DD);
    xm1 = *(const v4f*)(xb + (size_t)(l0 - 1) * DD);
  }

  // q/k/v segment (block-uniform: DBLK tiles never straddle split points)
  float* optr;
  int cstride, cl;
  if (d0 < 2048)      { optr = out;                           cstride = 2048; cl = c;        }
  else if (d0 < 4096) { optr = out + (size_t)BB * LL * 2048;  cstride = 2048; cl = c - 2048; }
  else                { optr = out + (size_t)BB * LL * 4096;  cstride = 4096; cl = c - 4096; }
  optr += (size_t)b * LL * cstride + cl;

  // --- async double-buffered pipeline over NTILE tiles ---
  {
    const float* g = xb + (size_t)l0 * DD;
    float* lb = smem;
    for (int row = 0; row < TLL; ++row)
      async_copy_b128(g + (size_t)row * DD, lb + row * DBLK + tid * 4);
  }
  for (int i = 0; i < NTILE; ++i) {
    if (i + 1 < NTILE) {
      const float* g = xb + (size_t)(l0 + (i + 1) * TLL) * DD;
      float* lb = smem + ((i + 1) & 1) * (TLL * DBLK);
      for (int row = 0; row < TLL; ++row)
        async_copy_b128(g + (size_t)row * DD, lb + row * DBLK + tid * 4);
      wait_async_prev();    // tile i has landed (async loads retire in order)
    } else {
      wait_async_all();
    }
    __syncthreads();        // all waves' rows visible in LDS

    const float* lb = smem + (i & 1) * (TLL * DBLK);
    for (int t = 0; t < TLL; ++t) {
      v4f cur = *(const v4f*)(lb + t * DBLK + tid * 4);
      v4f y;
      y.x = w0.x * xm3.x + w0.y * xm2.x + w0.z * xm1.x + w0.w * cur.x;
      y.y = w1.x * xm3.y + w1.y * xm2.y + w1.z * xm1.y + w1.w * cur.y;
      y.z = w2.x * xm3.z + w2.y * xm2.z + w2.z * xm1.z + w2.w * cur.z;
      y.w = w3.x * xm3.w + w3.y * xm2.w + w3.z * xm1.w + w3.w * cur.w;
      y.x = silu1(y.x); y.y = silu1(y.y); y.z = silu1(y.z); y.w = silu1(y.w);
      // write-once stream: non-temporal 16B store
      __builtin_nontemporal_store(
          y, (v4f*)(optr + (size_t)(l0 + i * TLL + t) * cstride));
      xm3 = xm2; xm2 = xm1; xm1 = cur;
    }
    __syncthreads();        // buffer free for reuse at i+2
  }
}

// new_cache[b, d, k] = x[b, L-4+k, d]  (pre-activation raw inputs)
__global__ __launch_bounds__(256) void cache_kernel(
    const float* __restrict__ x, float* __restrict__ out_cache) {
  int idx = blockIdx.x * blockDim.x + threadIdx.x;   // one per (b, d)
  int b = idx / DD;
  int d = idx % DD;
  const float* xb = x + ((size_t)b * LL + (LL - KK)) * DD + d;
  v4f v;
  v.x = xb[0];
  v.y = xb[(size_t)DD];
  v.z = xb[2 * (size_t)DD];
  v.w = xb[3 * (size_t)DD];
  *(v4f*)(out_cache + (size_t)idx * KK) = v;
}

extern "C" void kernel_launch(void* const* d_in, const int* in_sizes, int n_in,
                              void* d_out, int out_size, void* d_ws, size_t ws_size,
                              hipStream_t stream) {
  const float* x      = (const float*)d_in[0];
  const float* cache  = (const float*)d_in[1];
  const float* weight = (const float*)d_in[2];
  float* out = (float*)d_out;

  // q | k | v occupy the first B*L*D floats; new_cache follows.
  conv_silu_kernel<<<BB * (DD / DBLK) * LSPLIT, 256, 0, stream>>>(x, cache, weight, out);
  cache_kernel<<<(BB * DD) / 256, 256, 0, stream>>>(x, out + (size_t)BB * LL * DD);
}